// Spatial_Block_45492293599357
// MI455X (gfx1250) — compile-verified
//
#include <hip/hip_runtime.h>
#include <math.h>

#define KS   3
#define S_   8
#define CIN  8
#define COUT 16
#define NB   2
#define NV   1024
#define NT   32
#define E0   8192
#define EIDX 16384          // N*E0: offset of the "col" half of edge_index
#define NJ   (NB*NT)        // 64 graph copies
#define NTOT (NJ*NV)        // 65536 flattened node rows

typedef __attribute__((ext_vector_type(2))) float v2f;
typedef __attribute__((ext_vector_type(8))) float v8f;

__device__ __forceinline__ float eluf(float x) { return x > 0.f ? x : __expf(x) - 1.f; }

__device__ __forceinline__ v8f wmma4(v2f a, v2f b, v8f c) {
    return __builtin_amdgcn_wmma_f32_16x16x4_f32(false, a, false, b, (short)0, c, false, false);
}

// ---------------------------------------------------------------------------
// Kernel 0 (fused prep, one launch):
//  blocks [0,4096):  zero agg; build xf[(t*N+n)*V+v][c] = x[n][v][c][t]
//  blocks [4096,5120): per base edge, spline basis + combined weight
//      Wg[e][c][o] = sum_s basis[s] * weight[wi[s]][c][o]   (8x16 per edge)
// ---------------------------------------------------------------------------
__global__ void __launch_bounds__(256) k_prep(const float* __restrict__ x,
                                              const float* __restrict__ ea,
                                              const float* __restrict__ weight,
                                              float* __restrict__ agg,
                                              float* __restrict__ xf,
                                              float* __restrict__ Wg) {
    if (blockIdx.x < 4096) {
        int tid = blockIdx.x * 256 + threadIdx.x;   // covers NTOT*COUT = 1048576
        agg[tid] = 0.f;
        if (tid < NTOT * CIN) {
            int r = tid >> 3, c = tid & 7;
            int j = r >> 10, v = r & 1023;
            int t = j >> 1,  n = j & 1;
            xf[tid] = x[(((n * NV + v) * CIN + c) * NT) + t];
        }
        return;
    }
    int gtid = (blockIdx.x - 4096) * 256 + threadIdx.x;
    int e    = gtid >> 5;                           // 8192 edges, one wave each
    int lane = threadIdx.x & 31;

    float fr[3]; int fl[3];
#pragma unroll
    for (int d = 0; d < 3; ++d) {
        float v = ea[e * 3 + d] * (float)(KS - 1);
        float f = floorf(v);
        fl[d] = (int)f;
        fr[d] = v - f;
    }
    float basis[S_]; int wi[S_];
#pragma unroll
    for (int s = 0; s < S_; ++s) {
        float b = 1.f; int w = 0, stride = 1;
#pragma unroll
        for (int d = 0; d < 3; ++d) {
            int bit = (s >> d) & 1;
            int idx = fl[d] + bit; if (idx >= KS) idx -= KS;
            w += idx * stride; stride *= KS;
            b *= bit ? fr[d] : (1.f - fr[d]);
        }
        basis[s] = b; wi[s] = w;
    }
#pragma unroll
    for (int i = 0; i < 4; ++i) {                   // 128 outputs / 32 lanes
        int q = lane + 32 * i;                      // q = c*16+o
        float acc = 0.f;
#pragma unroll
        for (int s = 0; s < S_; ++s) acc += basis[s] * weight[wi[s] * 128 + q];
        Wg[e * 128 + q] = acc;
    }
}

// ---------------------------------------------------------------------------
// Kernel 1: edge GEMM via block-diagonal WMMA. No LDS, no barriers.
// Tile = 16 base edges. B(128x16) = stacked Wg of those edges: each lane's 64
// B-scalars are hoisted into VGPRs once and reused across all 8 j-iterations.
// A(16x128) is block-diagonal: lane (half,m) only ever contributes
// x[m][2h..2h+1] (even steps, k0==m) and x[m][4+2h..4+2h+1] (odd steps),
// fetched as two 8B global loads per j. D scattered with global_atomic_add_f32.
// ---------------------------------------------------------------------------
__global__ void __launch_bounds__(256) k_edge(const int* __restrict__ ei,
                                              const float* __restrict__ xf,
                                              const float* __restrict__ Wg,
                                              float* __restrict__ agg) {
    int tid  = threadIdx.x;
    int w    = tid >> 5;
    int lane = tid & 31;
    int half = lane >> 4;
    int m    = lane & 15;
    int e0   = blockIdx.x * 16;

    // B operand: 64 scalars per lane, loop-invariant -> registers
    const float* wgb = Wg + (size_t)e0 * 128 + m;
    float b0r[32], b1r[32];
#pragma unroll
    for (int t = 0; t < 32; ++t) {
        int kk0 = 4 * t + 2 * half;
        b0r[t] = wgb[kk0 * 16];          // B[kk0][n=m]
        b1r[t] = wgb[kk0 * 16 + 16];     // B[kk0+1][n=m]
    }
    // scatter rows for this lane's 8 D elements (M = r + 8*half)
    int rowb[8];
#pragma unroll
    for (int r = 0; r < 8; ++r) rowb[r] = ei[e0 + r + 8 * half];
    int colb = ei[EIDX + e0 + m];        // source node of edge m

    for (int jj = 0; jj < 8; ++jj) {     // each wave: 8 of the 64 graph copies
        int j   = w * 8 + jj;
        int col = colb + j * NV;
        const float* xrow = xf + (size_t)col * 8;
        v2f xlo = *(const v2f*)(xrow + 2 * half);        // x[m][2h],  x[m][2h+1]
        v2f xhi = *(const v2f*)(xrow + 4 + 2 * half);    // x[m][4+2h],x[m][4+2h+1]

        v8f acc = {0.f, 0.f, 0.f, 0.f, 0.f, 0.f, 0.f, 0.f};
#pragma unroll
        for (int t = 0; t < 32; ++t) {
            // kk0 = 4t+2h: k0 = t>>1, c0 = (t even) ? 2h : 4+2h
            bool on = ((t >> 1) == m);
            v2f xv  = (t & 1) ? xhi : xlo;
            v2f a, b;
            a.x = on ? xv.x : 0.f;
            a.y = on ? xv.y : 0.f;
            b.x = b0r[t];
            b.y = b1r[t];
            acc = wmma4(a, b, acc);
        }
#pragma unroll
        for (int r = 0; r < 8; ++r)
            atomicAdd(&agg[(size_t)(rowb[r] + j * NV) * COUT + m], acc[r]);
    }
}

// ---------------------------------------------------------------------------
// Kernel 2: per tile of 16 node rows (no LDS):
//   g   = elu(agg + xf@root + bias)     (agg enters as the WMMA C operand)
//   res = elu(xf@conv_w^T + conv_b)     (same A tile)
//   out[n][v][o][t] = elu(g + res)
// ---------------------------------------------------------------------------
__global__ void __launch_bounds__(256) k_final(const float* __restrict__ xf,
                                               const float* __restrict__ agg,
                                               const float* __restrict__ root,
                                               const float* __restrict__ bias,
                                               const float* __restrict__ convw,
                                               const float* __restrict__ convb,
                                               float* __restrict__ out) {
    int tid  = threadIdx.x;
    int w    = tid >> 5;
    int lane = tid & 31;
    int half = lane >> 4;
    int m    = lane & 15;
    int p0   = (blockIdx.x * 8 + w) * 16;            // 16 node rows per wave

    // A operand: lane's K-slices of row m (K=8 -> 2 wmma steps)
    const float* xrow = xf + (size_t)(p0 + m) * 8;
    v2f a0 = *(const v2f*)(xrow + 2 * half);         // step 0: kk = 2h, 2h+1
    v2f a1 = *(const v2f*)(xrow + 4 + 2 * half);     // step 1: kk = 4+2h, ...

    int kkA = 2 * half, kkB = 4 + 2 * half;
    v2f brA, brB;
    brA.x = root[kkA * 16 + m];       brA.y = root[(kkA + 1) * 16 + m];
    brB.x = root[kkB * 16 + m];       brB.y = root[(kkB + 1) * 16 + m];
    v2f bcA = *(const v2f*)(convw + m * CIN + kkA);  // conv_w[o=m][c=kkA..kkA+1]
    v2f bcB = *(const v2f*)(convw + m * CIN + kkB);

    // aggregation tile rides in as the C operand
    v8f cg;
#pragma unroll
    for (int r = 0; r < 8; ++r)
        cg[r] = agg[(size_t)(p0 + r + 8 * half) * COUT + m];
    v8f cc = {0.f, 0.f, 0.f, 0.f, 0.f, 0.f, 0.f, 0.f};

    cg = wmma4(a0, brA, cg);
    cg = wmma4(a1, brB, cg);
    cc = wmma4(a0, bcA, cc);
    cc = wmma4(a1, bcB, cc);

    float bo = bias[m], cbo = convb[m];
#pragma unroll
    for (int r = 0; r < 8; ++r) {
        int p = p0 + r + 8 * half;
        int j = p >> 10, v = p & 1023;
        int t = j >> 1,  n = j & 1;
        float g   = eluf(cg[r] + bo);
        float res = eluf(cc[r] + cbo);
        out[(((size_t)(n * NV + v) * COUT + m) * NT) + t] = eluf(g + res);
    }
}

// ---------------------------------------------------------------------------
extern "C" void kernel_launch(void* const* d_in, const int* in_sizes, int n_in,
                              void* d_out, int out_size, void* d_ws, size_t ws_size,
                              hipStream_t stream) {
    const float* x      = (const float*)d_in[0];
    const int*   ei     = (const int*)  d_in[1];
    const float* ea     = (const float*)d_in[2];
    const float* weight = (const float*)d_in[3];
    const float* root   = (const float*)d_in[4];
    const float* bias   = (const float*)d_in[5];
    const float* convw  = (const float*)d_in[6];
    const float* convb  = (const float*)d_in[7];
    float* out = (float*)d_out;

    char* ws = (char*)d_ws;
    float* Wg  = (float*)ws;                                          // 8192*128*4 = 4 MB
    float* agg = (float*)(ws + (size_t)E0 * 128 * 4);                 // 65536*16*4 = 4 MB
    float* xf  = (float*)(ws + (size_t)E0 * 128 * 4
                             + (size_t)NTOT * COUT * 4);              // 65536*8*4 = 2 MB

    k_prep <<<5120, 256, 0, stream>>>(x, ea, weight, agg, xf, Wg);
    k_edge <<< 512, 256, 0, stream>>>(ei, xf, Wg, agg);
    k_final<<< 512, 256, 0, stream>>>(xf, agg, root, bias, convw, convb, out);
}